// _LearnedEntityEmbedding_13348758356140
// MI455X (gfx1250) — compile-verified
//
#include <hip/hip_runtime.h>

#define NUM_NUM 32
#define NUM_CAT 32
#define CARD    128
#define OUTF    32
#define BATCH   16384
#define LDX     (NUM_NUM + NUM_CAT * CARD)   // 4128 floats per input row
#define LDO     (NUM_NUM + NUM_CAT * OUTF)   // 1056 floats per output row
#define WPAD    (CARD + 4)                   // LDS row stride (floats) -> conflict-free b64 reads

typedef __attribute__((ext_vector_type(2))) float v2f;
typedef __attribute__((ext_vector_type(4))) float v4f;
typedef __attribute__((ext_vector_type(8))) float v8f;

// One block: 64 batch rows (4 waves x 16 rows) for one categorical feature f.
// Each wave computes a 16x32 tile: D(16x32) = A(16x128) * W_f(128x32) + bias_f
// using V_WMMA_F32_16X16X4_F32, accumulating over K in steps of 4.
__global__ __launch_bounds__(128) void emb_gemm_wmma(const float* __restrict__ x,
                                                     const float* __restrict__ W,
                                                     const float* __restrict__ bias,
                                                     float* __restrict__ out) {
  // W_f transposed into LDS: Wl[n * WPAD + k] = W[f][k][n]. 32*132*4B ~= 16.9 KB.
  __shared__ float Wl[OUTF * WPAD];
  __shared__ float bl[OUTF];

  const int f   = blockIdx.y;
  const int tid = threadIdx.x;

  // ---- stage W[f] (coalesced global reads; one-time strided LDS writes) ----
  const float* __restrict__ Wf = W + (size_t)f * (CARD * OUTF);
#pragma unroll
  for (int i = 0; i < (CARD * OUTF) / 128; ++i) {
    int idx = tid + i * 128;          // linear over k*32 + n
    int k = idx >> 5;
    int n = idx & 31;
    Wl[n * WPAD + k] = Wf[idx];
  }
  if (tid < OUTF) bl[tid] = bias[f * OUTF + tid];
  __syncthreads();

  const int wave = tid >> 5;          // 0..3 -> M sub-tile
  const int lane = tid & 31;
  const int half = lane >> 4;         // 0: K+{0,1} group, 1: K+{2,3} group
  const int l16  = lane & 15;

  const int m_base = blockIdx.x * 64 + wave * 16;

  // A-operand base: lane (l16, half) reads rows m_base+l16, cols k0 + 2*half (+0,+1)
  const float* __restrict__ Arow =
      x + (size_t)(m_base + l16) * LDX + NUM_NUM + f * CARD + half * 2;

  // B-operand bases in LDS (transposed): row n, cols k0 + 2*half (+0,+1) -> one b64
  const float* __restrict__ B0 = &Wl[l16 * WPAD + half * 2];          // n-tile 0
  const float* __restrict__ B1 = &Wl[(16 + l16) * WPAD + half * 2];   // n-tile 1

  // Accumulators pre-loaded with bias (bias depends only on N = l16 / 16+l16).
  const float bn0 = bl[l16];
  const float bn1 = bl[16 + l16];
  v8f c0, c1;
#pragma unroll
  for (int r = 0; r < 8; ++r) { c0[r] = bn0; c1[r] = bn1; }

#pragma unroll 8
  for (int k0 = 0; k0 < CARD; k0 += 4) {
    v2f a  = *(const v2f*)(Arow + k0);   // global_load_b64, 8B aligned
    v2f b0 = *(const v2f*)(B0 + k0);     // ds_load_b64, conflict-free
    v2f b1 = *(const v2f*)(B1 + k0);
    c0 = __builtin_amdgcn_wmma_f32_16x16x4_f32(false, a, false, b0, (short)0, c0,
                                               false, false);
    c1 = __builtin_amdgcn_wmma_f32_16x16x4_f32(false, a, false, b1, (short)0, c1,
                                               false, false);
  }

  // D layout: VGPR r -> row m_base + r + 8*half, col = NUM_NUM + f*32 + ntile*16 + l16
  float* __restrict__ orow =
      out + (size_t)(m_base + half * 8) * LDO + NUM_NUM + f * OUTF + l16;
#pragma unroll
  for (int r = 0; r < 8; ++r) {
    orow[(size_t)r * LDO]      = c0[r];
    orow[(size_t)r * LDO + 16] = c1[r];
  }
}

// Pass-through of the 32 numeric columns: out[m, 0:32] = x[m, 0:32], float4 wide.
__global__ __launch_bounds__(256) void copy_num(const float* __restrict__ x,
                                                float* __restrict__ out) {
  int t = blockIdx.x * blockDim.x + threadIdx.x;   // BATCH * 8 threads
  int m = t >> 3;
  int c = (t & 7) * 4;
  v4f v = *(const v4f*)(x + (size_t)m * LDX + c);
  *(v4f*)(out + (size_t)m * LDO + c) = v;
}

extern "C" void kernel_launch(void* const* d_in, const int* in_sizes, int n_in,
                              void* d_out, int out_size, void* d_ws, size_t ws_size,
                              hipStream_t stream) {
  const float* x = (const float*)d_in[0];   // (16384, 4128)
  const float* W = (const float*)d_in[1];   // (32, 128, 32)
  const float* b = (const float*)d_in[2];   // (32, 32)
  float* out = (float*)d_out;               // (16384, 1056)

  dim3 grid(BATCH / 64, NUM_CAT);           // (256, 32)
  emb_gemm_wmma<<<grid, 128, 0, stream>>>(x, W, b, out);

  copy_num<<<(BATCH * 8) / 256, 256, 0, stream>>>(x, out);
}